// QAvgPool2d_60344290508885
// MI455X (gfx1250) — compile-verified
//
#include <hip/hip_runtime.h>
#include <stdint.h>

// QAvgPool2d: (32,256,112,112) f32 -> (32,256,56,56) f32, 2x2/s2 avg pool with
// fp16 (E5M10) RNE quantization after every partial-sum add and after /4.
// Pure HBM-bound streaming kernel (~514 MB moved, ~22 us floor at 23.3 TB/s).

typedef float v4f __attribute__((ext_vector_type(4)));
typedef float v2f __attribute__((ext_vector_type(2)));
typedef int   v4i __attribute__((ext_vector_type(4)));

#define HAS_ASYNC_LDS __has_builtin(__builtin_amdgcn_global_load_async_to_lds_b128)

#if __has_builtin(__builtin_amdgcn_s_wait_asynccnt)
#define WAIT_ASYNCCNT(n) do { __builtin_amdgcn_s_wait_asynccnt(n); asm volatile("" ::: "memory"); } while (0)
#else
#define WAIT_ASYNCCNT(n) asm volatile("s_wait_asynccnt %0" :: "n"(n) : "memory")
#endif

// Problem constants
constexpr unsigned Wd   = 112;               // input width
constexpr unsigned PLIN = 112u * 112u;       // 12544 input elems / plane
constexpr unsigned PLOUT = 56u * 56u;        // 3136 output elems / plane
constexpr unsigned UNITS_PER_ROW   = 28;     // OW/2 (each unit -> 2 outputs)
constexpr unsigned UNITS_PER_PLANE = 56u * UNITS_PER_ROW;   // 1568
constexpr unsigned PLANES = 32u * 256u;      // 8192
constexpr unsigned UNITS_TOTAL = PLANES * UNITS_PER_PLANE;  // 12,845,056
constexpr int STAGES = 8;
constexpr unsigned BLOCKS = UNITS_TOTAL / (256u * STAGES);  // 6272 exactly

static_assert(UNITS_TOTAL == BLOCKS * 256u * STAGES, "exact tiling");

// fp16 E5M10 RNE quantization via hardware cvt pair (denorms preserved;
// saturation irrelevant for N(0,1) data, |sum| << 65504).
__device__ __forceinline__ float q16(float v) { return (float)(_Float16)v; }

__device__ __forceinline__ float pool1(float a, float b, float c, float d) {
    float s = q16(a);        // q(0 + p00)
    s = q16(s + b);          // q(s + p01)
    s = q16(s + c);          // q(s + p10)
    s = q16(s + d);          // q(s + p11)
    return q16(s * 0.25f);   // q(s / 4)  (exact scale; re-rounds subnormals)
}

__device__ __forceinline__ void unit_addr(unsigned u, unsigned& in_off, unsigned& out_off) {
    unsigned p  = u % UNITS_PER_ROW;           // column-pair index [0,28)
    unsigned t  = u / UNITS_PER_ROW;
    unsigned r  = t % 56u;                     // output row
    unsigned pl = t / 56u;                     // (b,c) plane
    in_off  = pl * PLIN  + r * (2u * Wd) + p * 4u;   // float4-aligned
    out_off = pl * PLOUT + r * 56u       + p * 2u;   // float2-aligned
}

#if HAS_ASYNC_LDS
__device__ __forceinline__ void async_copy_b128(const float* gsrc, const float* lds_dst) {
    // generic->AS1 / generic->AS3 via integer round-trip (LDS generic low 32
    // bits are the LDS offset on CDNA5). Builtin signature (from diagnostics):
    // (v4i AS1*, v4i AS3*, imm offset, imm cpol).
    auto g = (__attribute__((address_space(1))) v4i*)(uintptr_t)gsrc;
    auto l = (__attribute__((address_space(3))) v4i*)(uint32_t)(uintptr_t)lds_dst;
    __builtin_amdgcn_global_load_async_to_lds_b128(g, l, 0, 0);
}
#endif

__global__ __launch_bounds__(256) void qavgpool2x2_kernel(const float* __restrict__ x,
                                                          float* __restrict__ y) {
    const unsigned tid  = threadIdx.x;
    const unsigned base = blockIdx.x * (unsigned)STAGES;

#if HAS_ASYNC_LDS
    // Double-buffered staging: [buf][row 0/1][lane][4 floats] = 16 KB.
    // Lane-major 16B slots -> conflict-free ds_load_b128 readback.
    __shared__ __align__(16) float lds[2][2][256][4];

    auto issue = [&](int s, int buf) {
        unsigned u, in_off, out_off;
        u = (base + (unsigned)s) * 256u + tid;
        unit_addr(u, in_off, out_off);
        async_copy_b128(x + in_off,      &lds[buf][0][tid][0]);   // input row 2*oh
        async_copy_b128(x + in_off + Wd, &lds[buf][1][tid][0]);   // input row 2*oh+1
    };

    issue(0, 0);
    for (int s = 0; s < STAGES; ++s) {
        const int buf = s & 1;
        if (s + 1 < STAGES) {
            issue(s + 1, buf ^ 1);
            WAIT_ASYNCCNT(2);   // in-order completion: stage s's 2 loads retired
        } else {
            WAIT_ASYNCCNT(0);
        }
        v4f r0 = *(const v4f*)&lds[buf][0][tid][0];
        v4f r1 = *(const v4f*)&lds[buf][1][tid][0];

        unsigned u = (base + (unsigned)s) * 256u + tid;
        unsigned in_off, out_off;
        unit_addr(u, in_off, out_off);

        v2f o;
        o.x = pool1(r0.x, r0.y, r1.x, r1.y);
        o.y = pool1(r0.z, r0.w, r1.z, r1.w);
        __builtin_nontemporal_store(o, (v2f*)(y + out_off));
    }
#else
    // Fallback: direct non-temporal b128 streaming loads.
    for (int s = 0; s < STAGES; ++s) {
        unsigned u = (base + (unsigned)s) * 256u + tid;
        unsigned in_off, out_off;
        unit_addr(u, in_off, out_off);

        v4f r0 = __builtin_nontemporal_load((const v4f*)(x + in_off));
        v4f r1 = __builtin_nontemporal_load((const v4f*)(x + in_off + Wd));

        v2f o;
        o.x = pool1(r0.x, r0.y, r1.x, r1.y);
        o.y = pool1(r0.z, r0.w, r1.z, r1.w);
        __builtin_nontemporal_store(o, (v2f*)(y + out_off));
    }
#endif
}

extern "C" void kernel_launch(void* const* d_in, const int* in_sizes, int n_in,
                              void* d_out, int out_size, void* d_ws, size_t ws_size,
                              hipStream_t stream) {
    (void)in_sizes; (void)n_in; (void)d_ws; (void)ws_size; (void)out_size;
    const float* x = (const float*)d_in[0];
    float* y = (float*)d_out;
    qavgpool2x2_kernel<<<dim3(BLOCKS), dim3(256), 0, stream>>>(x, y);
}